// HeteroGNN_60120952209794
// MI455X (gfx1250) — compile-verified
//
#include <hip/hip_runtime.h>

// ---------------------------------------------------------------------------
// HeteroGNN (2-layer hetero GraphSAGE) for MI455X / gfx1250.
// GEMMs use V_WMMA_F32_16X16X4_F32 (fp32 tensor path, matches fp32 reference).
// Edge aggregation uses global_atomic_add_f32 (L2-resident: feature tables
// are ~13 MB total << 192 MB L2).
// ---------------------------------------------------------------------------

typedef __attribute__((ext_vector_type(2))) float v2f;
typedef __attribute__((ext_vector_type(8))) float v8f;

static constexpr int HD     = 128;    // hidden dim
static constexpr int KIN    = 512;    // input feature dim
static constexpr int NGENE  = 20000;
static constexpr int NDRUG  = 5000;
static constexpr int LDA_IN = 516;    // padded LDS stride for 512-wide A tile
static constexpr int LDB_H  = 132;    // padded LDS stride for 128-wide A tile

__device__ __forceinline__ v8f wmma4(v2f a, v2f b, v8f c) {
  // D = A(16x4 f32) x B(4x16 f32) + C(16x16 f32)
  return __builtin_amdgcn_wmma_f32_16x16x4_f32(
      /*neg_a=*/false, a, /*neg_b=*/false, b,
      /*c_mod=*/(short)0, c, /*reuse_a=*/false, /*reuse_b=*/false);
}

// C[N,128] = A[N,512] @ W[512,128] + bias.  Block = 256 threads = 8 waves,
// each wave owns one 16-col tile; block covers 16 rows x 128 cols.
__global__ __launch_bounds__(256) void gemm_in_kernel(
    const float* __restrict__ A, const float* __restrict__ W,
    const float* __restrict__ bias, float* __restrict__ C, int N)
{
  __shared__ float As[16 * LDA_IN];
  const int tid = threadIdx.x;
  const int rowBase = blockIdx.x * 16;

  // Stage A tile [16 x 512] (clamped rows), coalesced float4 loads.
  #pragma unroll
  for (int i = 0; i < 8; ++i) {
    int idx = tid + i * 256;          // float4 index in [0, 2048)
    int r   = idx >> 7;               // 128 float4 per row
    int c4  = idx & 127;
    int gr  = rowBase + r; if (gr > N - 1) gr = N - 1;
    const float4 v = ((const float4*)(A + (size_t)gr * KIN))[c4];
    float* p = &As[r * LDA_IN + c4 * 4];
    p[0] = v.x; p[1] = v.y; p[2] = v.z; p[3] = v.w;
  }
  __syncthreads();

  const int wave = tid >> 5;
  const int lane = tid & 31;
  const int m    = lane & 15;
  const int koff = (lane >> 4) << 1;      // 0 or 2 (A/B K split across half-waves)
  const int col  = wave * 16 + m;

  v8f acc = {0.f,0.f,0.f,0.f,0.f,0.f,0.f,0.f};
  const float* As_row = &As[m * LDA_IN + koff];
  const float* Wp     = W + (size_t)koff * HD + col;

  #pragma unroll 4
  for (int k = 0; k < KIN; k += 4) {
    v2f a = *(const v2f*)(As_row + k);
    v2f b;
    b[0] = Wp[(size_t)k * HD];
    b[1] = Wp[(size_t)(k + 1) * HD];
    acc = wmma4(a, b, acc);
  }

  const float bv  = bias[col];
  const int mrow  = (lane >> 4) << 3;     // 0 or 8
  #pragma unroll
  for (int r = 0; r < 8; ++r) {
    int row = rowBase + r + mrow;
    if (row < N) C[(size_t)row * HD + col] = acc[r] + bv;
  }
}

// out[N,128] += (Agg/max(cnt,1)) @ Wl + bl + Xd @ Wr
__global__ __launch_bounds__(256) void sage_gemm_kernel(
    const float* __restrict__ Agg, const float* __restrict__ cnt,
    const float* __restrict__ Wl,  const float* __restrict__ Xd,
    const float* __restrict__ Wr,  const float* __restrict__ bias,
    float* __restrict__ out, int N)
{
  __shared__ float s0[16 * LDB_H];
  __shared__ float s1[16 * LDB_H];
  const int tid = threadIdx.x;
  const int rowBase = blockIdx.x * 16;

  // Stage both 16x128 tiles; fold mean division into the staging of Agg.
  #pragma unroll
  for (int i = 0; i < 2; ++i) {
    int idx = tid + i * 256;          // float4 index in [0, 512)
    int r   = idx >> 5;               // 32 float4 per row
    int c4  = idx & 31;
    int gr  = rowBase + r; if (gr > N - 1) gr = N - 1;
    float inv = 1.0f / fmaxf(cnt[gr], 1.0f);
    float4 v0 = ((const float4*)(Agg + (size_t)gr * HD))[c4];
    float4 v1 = ((const float4*)(Xd  + (size_t)gr * HD))[c4];
    float* p0 = &s0[r * LDB_H + c4 * 4];
    p0[0] = v0.x * inv; p0[1] = v0.y * inv; p0[2] = v0.z * inv; p0[3] = v0.w * inv;
    float* p1 = &s1[r * LDB_H + c4 * 4];
    p1[0] = v1.x; p1[1] = v1.y; p1[2] = v1.z; p1[3] = v1.w;
  }
  __syncthreads();

  const int wave = tid >> 5;
  const int lane = tid & 31;
  const int m    = lane & 15;
  const int koff = (lane >> 4) << 1;
  const int col  = wave * 16 + m;

  v8f acc = {0.f,0.f,0.f,0.f,0.f,0.f,0.f,0.f};
  const float* r0  = &s0[m * LDB_H + koff];
  const float* r1  = &s1[m * LDB_H + koff];
  const float* Wlp = Wl + (size_t)koff * HD + col;
  const float* Wrp = Wr + (size_t)koff * HD + col;

  #pragma unroll 4
  for (int k = 0; k < HD; k += 4) {
    v2f a = *(const v2f*)(r0 + k);
    v2f b; b[0] = Wlp[(size_t)k * HD]; b[1] = Wlp[(size_t)(k + 1) * HD];
    acc = wmma4(a, b, acc);
  }
  #pragma unroll 4
  for (int k = 0; k < HD; k += 4) {
    v2f a = *(const v2f*)(r1 + k);
    v2f b; b[0] = Wrp[(size_t)k * HD]; b[1] = Wrp[(size_t)(k + 1) * HD];
    acc = wmma4(a, b, acc);
  }

  const float bv = bias[col];
  const int mrow = (lane >> 4) << 3;
  #pragma unroll
  for (int r = 0; r < 8; ++r) {
    int row = rowBase + r + mrow;
    if (row < N) {
      size_t o = (size_t)row * HD + col;
      out[o] = out[o] + acc[r] + bv;
    }
  }
}

// One wave per edge: gather 128 floats from X[src], atomic-add into agg[dst].
__global__ __launch_bounds__(256) void scatter_mean_kernel(
    const float* __restrict__ X, const int* __restrict__ src,
    const int* __restrict__ dst, float* __restrict__ agg,
    float* __restrict__ cnt, int E)
{
  int t = blockIdx.x * blockDim.x + threadIdx.x;
  int e = t >> 5, lane = t & 31;
  if (e >= E) return;
  int s = src[e], d = dst[e];
  const float* xs = X + (size_t)s * HD;
  float* ap = agg + (size_t)d * HD;
  #pragma unroll
  for (int i = 0; i < 4; ++i)
    atomicAdd(ap + lane + 32 * i, xs[lane + 32 * i]);
  if (lane == 0) atomicAdd(cnt + d, 1.0f);
}

__global__ __launch_bounds__(256) void relu_kernel(
    const float* __restrict__ in, float* __restrict__ out, int n)
{
  int i = blockIdx.x * blockDim.x + threadIdx.x;
  if (i < n) out[i] = fmaxf(in[i], 0.0f);
}

extern "C" void kernel_launch(void* const* d_in, const int* in_sizes, int n_in,
                              void* d_out, int out_size, void* d_ws, size_t ws_size,
                              hipStream_t stream)
{
  (void)n_in; (void)out_size; (void)ws_size;

  const float* x_gene = (const float*)d_in[0];
  const float* x_drug = (const float*)d_in[1];

  // Relations in JAX pytree (sorted-key) param order:
  //   DDS, DTI_dg, DTI_gd, GSEA, PCC, PPI
  // Input-dict edge indices: ppi(2,3) gsea(4,5) pcc(6,7) dds(8,9)
  //                          dti_dg(10,11) dti_gd(12,13)
  struct Rel { int src_in, dst_in, st, dt; };   // st/dt: 0=gene, 1=drug
  const Rel rels[6] = {
    {8,  9,  1, 1},   // DDS    drug->drug
    {10, 11, 1, 0},   // DTI_dg drug->gene
    {12, 13, 0, 1},   // DTI_gd gene->drug
    {4,  5,  0, 0},   // GSEA   gene->gene
    {6,  7,  0, 0},   // PCC    gene->gene
    {2,  3,  0, 0},   // PPI    gene->gene
  };

  // Params (leaves, sorted keys): drug_in(W,b), gene_in(W,b),
  // then layers[0..1] x {DDS,DTI_dg,DTI_gd,GSEA,PCC,PPI} x (Wl, bl, Wr)
  const float* drug_in_W = (const float*)d_in[14];
  const float* drug_in_b = (const float*)d_in[15];
  const float* gene_in_W = (const float*)d_in[16];
  const float* gene_in_b = (const float*)d_in[17];

  // Workspace layout (floats)
  float* ws  = (float*)d_ws;
  float* hg  = ws;                         // 20000*128 gene features
  float* hd  = hg  + (size_t)NGENE * HD;   // 5000*128  drug features
  float* og  = hd  + (size_t)NDRUG * HD;   // gene pre-activation accumulator
  float* od  = og  + (size_t)NGENE * HD;   // drug pre-activation accumulator
  float* agg = od  + (size_t)NDRUG * HD;   // shared agg scratch (max 20000*128)
  float* cnt = agg + (size_t)NGENE * HD;   // shared count scratch (max 20000)

  dim3 blk(256);

  // Input projections (no ReLU)
  gemm_in_kernel<<<dim3((NGENE + 15) / 16), blk, 0, stream>>>(
      x_gene, gene_in_W, gene_in_b, hg, NGENE);
  gemm_in_kernel<<<dim3((NDRUG + 15) / 16), blk, 0, stream>>>(
      x_drug, drug_in_W, drug_in_b, hd, NDRUG);

  float* h[2] = {hg, hd};
  float* o[2] = {og, od};
  const int nsz[2] = {NGENE, NDRUG};

  for (int l = 0; l < 2; ++l) {
    hipMemsetAsync(og, 0, (size_t)NGENE * HD * sizeof(float), stream);
    hipMemsetAsync(od, 0, (size_t)NDRUG * HD * sizeof(float), stream);

    for (int r = 0; r < 6; ++r) {
      const Rel& R = rels[r];
      const int nd = nsz[R.dt];
      const int E  = in_sizes[R.src_in];

      hipMemsetAsync(agg, 0, (size_t)nd * HD * sizeof(float), stream);
      hipMemsetAsync(cnt, 0, (size_t)nd * sizeof(float), stream);

      long long nth = (long long)E * 32;
      scatter_mean_kernel<<<dim3((unsigned)((nth + 255) / 256)), blk, 0, stream>>>(
          h[R.st], (const int*)d_in[R.src_in], (const int*)d_in[R.dst_in],
          agg, cnt, E);

      const float* Wl = (const float*)d_in[18 + 18 * l + 3 * r + 0];
      const float* bl = (const float*)d_in[18 + 18 * l + 3 * r + 1];
      const float* Wr = (const float*)d_in[18 + 18 * l + 3 * r + 2];
      sage_gemm_kernel<<<dim3((nd + 15) / 16), blk, 0, stream>>>(
          agg, cnt, Wl, h[R.dt], Wr, bl, o[R.dt], nd);
    }

    if (l == 0) {
      relu_kernel<<<dim3((NGENE * HD + 255) / 256), blk, 0, stream>>>(og, hg, NGENE * HD);
      relu_kernel<<<dim3((NDRUG * HD + 255) / 256), blk, 0, stream>>>(od, hd, NDRUG * HD);
    } else {
      float* outg = (float*)d_out;
      float* outd = outg + (size_t)NGENE * HD;
      relu_kernel<<<dim3((NGENE * HD + 255) / 256), blk, 0, stream>>>(og, outg, NGENE * HD);
      relu_kernel<<<dim3((NDRUG * HD + 255) / 256), blk, 0, stream>>>(od, outd, NDRUG * HD);
    }
  }
}